// K2CTCLoss_43336220016736
// MI455X (gfx1250) — compile-verified
//
#include <hip/hip_runtime.h>
#include <cstdint>
#include <cstddef>

// CTC forward (alpha) recursion, one workgroup per utterance.
// N=32, T=2048, C=512, S=256, L=2S+1=513.
//
// CDNA5 data path: GLOBAL_LOAD_ASYNC_TO_LDS_B128 with a 14-deep LDS row ring
// (prefetch distance D=12 timesteps, ~2-5K cycles of HBM/L2 latency cover),
// drained with S_WAIT_ASYNCCNT. Past the stream end, loaders issue clamped
// duplicate loads so the in-flight depth stays exactly D and the immediate
// wait value remains correct for every iteration.

#define CTC_NEG_INF (-1e30f)
#define CTC_C    512          // class count (fixed by the reference problem)
#define CTC_LMAX 516          // padded max extended-label length (L = 513)
#define CTC_D    12           // prefetch distance in rows (timesteps)
#define CTC_R    (CTC_D + 2)  // row ring depth: reuse separated by >=1 barrier

// ---- CDNA5 async global->LDS copy of 16 bytes per lane (GV addressing) ----
__device__ __forceinline__ void async_ld_b128(uint32_t lds_addr, const float* gaddr) {
  // vdst = per-lane LDS byte address, vaddr = 64-bit global address
  asm volatile("global_load_async_to_lds_b128 %0, %1, off"
               :: "v"(lds_addr), "v"(gaddr)
               : "memory");
}
__device__ __forceinline__ void wait_async_leD() {
  asm volatile("s_wait_asynccnt 0xc" ::: "memory");   // <= CTC_D (12)
}

__device__ __forceinline__ uint32_t lds_addr_of(const void* p) {
  // generic pointer to LDS: low 32 bits are the wave-relative LDS byte address
  return (uint32_t)(uintptr_t)p;
}

__global__ __launch_bounds__(544, 1) void ctc_alpha_kernel(
    const float* __restrict__ log_probs,   // (N, T, C)
    const int*   __restrict__ targets,     // (N, S)
    const int*   __restrict__ input_len,   // (N,)
    const int*   __restrict__ target_len,  // (N,)
    float*       __restrict__ out,         // (N,)
    int T, int S)
{
  constexpr int C = CTC_C;
  const int n = blockIdx.x;
  const int l = threadIdx.x;
  const int L = 2 * S + 1;

  const bool active = (l < L);
  const bool loader = (l < (C / 4));   // threads 0..127 = waves 0..3, full waves

  __shared__ float rb[CTC_R][C];       // log-prob row ring (14 x 2KB = 28KB)
  __shared__ float ab[2][CTC_LMAX];    // ping-pong alpha

  const float* rowbase = log_probs + (size_t)n * (size_t)T * (size_t)C;

  // ---- per-state constants: extended label and skip-transition legality ----
  int  ext_l = 0;           // even states are blank (class 0)
  bool allow_skip = false;  // s-2 -> s legal only for non-repeated labels, odd l>=3
  if (active && (l & 1)) {
    ext_l = targets[n * S + (l >> 1)];
    if (l >= 3) allow_skip = (ext_l != targets[n * S + ((l - 2) >> 1)]);
  }

  // ---- prologue: fill the pipeline with rows 0..D (13 async issues/wave) ----
  if (loader) {
    const uint32_t lds_col = (uint32_t)(l * 16);   // 16B per lane
    for (int r = 0; r <= CTC_D; ++r) {
      const int rr = (r < T) ? r : (T - 1);        // clamp for tiny T
      async_ld_b128(lds_addr_of(&rb[r][0]) + lds_col,
                    rowbase + (size_t)rr * C + l * 4);
    }
    wait_async_leD();        // D+1 issued, wait <=D  =>  row 0 has landed
  }
  __syncthreads();

  float alpha = CTC_NEG_INF;
  if (active && l < 2) alpha = rb[0][ext_l];   // alpha0: states 0 (blank) and 1
  const int len_n = input_len[n];

  // ---- main recursion: one barrier per timestep, constant-depth pipeline ----
  int slot_cur = 1 % CTC_R;             // ring slot holding row t (t starts at 1)
  int slot_pre = (1 + CTC_D) % CTC_R;   // ring slot receiving row t+D
  for (int t = 1; t < T; ++t) {
    if (loader) {
      // Issue row t+D (clamped duplicates past the end keep in-flight == D,
      // so the immediate wait below always implies row t is complete).
      int rr = t + CTC_D; if (rr >= T) rr = T - 1;
      async_ld_b128(lds_addr_of(&rb[slot_pre][0]) + (uint32_t)(l * 16),
                    rowbase + (size_t)rr * C + l * 4);
      wait_async_leD();      // rows <= t complete; t+1..t+D may be in flight
    }
    if (active) ab[t & 1][l] = alpha;
    __syncthreads();         // alpha + async row both visible block-wide

    if (active) {
      const float e  = rb[slot_cur][ext_l];
      const int  lm1 = (l > 0) ? l - 1 : 0;
      const int  lm2 = (l > 1) ? l - 2 : 0;
      const float a0 = alpha;
      const float a1 = (l >= 1)   ? ab[t & 1][lm1] : CTC_NEG_INF;
      const float a2 = allow_skip ? ab[t & 1][lm2] : CTC_NEG_INF;
      const float m  = fmaxf(a0, fmaxf(a1, a2));
      const float s  = __expf(a0 - m) + __expf(a1 - m) + __expf(a2 - m);
      const float nv = m + __logf(s) + e;
      if (t < len_n) alpha = nv;  // freeze past utterance end
    }
    slot_cur = (slot_cur + 1 == CTC_R) ? 0 : slot_cur + 1;
    slot_pre = (slot_pre + 1 == CTC_R) ? 0 : slot_pre + 1;
  }

  // ---- epilogue: loss = -logsumexp(alpha[2*tl], alpha[2*tl-1]) ----
  __syncthreads();
  if (active) ab[0][l] = alpha;
  __syncthreads();
  if (l == 0) {
    const int   tl = target_len[n];
    const float x  = ab[0][2 * tl];
    const float y  = ab[0][2 * tl - 1];
    const float m  = fmaxf(x, y);
    out[n] = -(m + __logf(__expf(x - m) + __expf(y - m)));
  }
}

extern "C" void kernel_launch(void* const* d_in, const int* in_sizes, int n_in,
                              void* d_out, int out_size, void* d_ws, size_t ws_size,
                              hipStream_t stream) {
  (void)n_in; (void)d_ws; (void)ws_size; (void)out_size;
  const float* log_probs  = (const float*)d_in[0];
  const int*   targets    = (const int*)d_in[1];
  const int*   input_len  = (const int*)d_in[2];
  const int*   target_len = (const int*)d_in[3];

  const int N = in_sizes[2];                     // 32
  const int S = in_sizes[1] / N;                 // 256
  const int T = in_sizes[0] / (N * CTC_C);       // 2048
  const int L = 2 * S + 1;                       // 513
  const int block = ((L + 31) / 32) * 32;        // 544 = 17 wave32s

  ctc_alpha_kernel<<<N, block, 0, stream>>>(log_probs, targets, input_len,
                                            target_len, (float*)d_out, T, S);
}